// SceneGCN_6262062317721
// MI455X (gfx1250) — compile-verified
//
#include <hip/hip_runtime.h>
#include <math.h>

// ---------------- problem constants ----------------
#define NN   50000      // nodes
#define EE   800000     // raw edges
#define ET   850000     // edges + self loops
#define NG   128        // graphs

typedef __attribute__((ext_vector_type(2))) float v2f;
typedef __attribute__((ext_vector_type(8))) float v8f;

// ---------------- helpers ----------------
__device__ __forceinline__ void edge_sd(const int* __restrict__ ei, int e, int& s, int& d) {
  if (e < EE) { s = ei[e]; d = ei[EE + e]; }
  else        { s = d = e - EE; }          // self loop
}

__device__ __forceinline__ float atomicMaxF(float* addr, float val) {
  if (val >= 0.f)
    return __int_as_float(atomicMax((int*)addr, __float_as_int(val)));
  else
    return __uint_as_float(atomicMin((unsigned int*)addr, __float_as_uint(val)));
}

// ---------------- fp32 WMMA GEMM: C[M,N] (+)= A[M,K] @ B[K,N] ----------------
// One wave per 16x16 output tile; K in chunks of 4 via v_wmma_f32_16x16x4_f32.
// A fragment (16x4 f32): lanes 0-15 hold K={0,1} in 2 VGPRs, lanes 16-31 K={2,3}.
// B fragment (4x16):     lane&15 = N column, same K split.
// C/D (16x16 f32, 8 VGPRs): M = r + 8*(lane>=16), N = lane&15.
// M multiple of 16 (50000 = 3125*16), N multiple of 16, K even (14/64/256).
__global__ void gemm_wmma_f32(const float* __restrict__ A, const float* __restrict__ B,
                              float* __restrict__ C, int Mdim, int Ndim, int Kdim,
                              int accumulate) {
  const int wid  = (blockIdx.x * blockDim.x + threadIdx.x) >> 5;
  const int lane = threadIdx.x & 31;
  const int nt   = Ndim >> 4;
  const int tm   = wid / nt;
  const int tn   = wid % nt;
  if (tm * 16 >= Mdim) return;                 // wave-uniform
  const int half = lane >> 4;                  // 0: K pair {0,1}, 1: {2,3}
  const int l    = lane & 15;
  const int arow = tm * 16 + l;

  float* Cp = C + (size_t)(tm*16 + half*8) * Ndim + tn*16 + l;

  v8f c = {};
  if (accumulate) {
#pragma unroll
    for (int r = 0; r < 8; ++r) c[r] = Cp[(size_t)r * Ndim];
  }

  // ---- main loop: complete chunks of 4, no guards, pointer-increment addressing
  const int kfull = Kdim >> 2;
  const float* Ap = A + (size_t)arow * Kdim + half * 2;            // even -> 8B aligned
  const float* Bp = B + (size_t)(half * 2) * Ndim + tn*16 + l;
  for (int kk = 0; kk < kfull; ++kk) {
    const float2 av = *(const float2*)Ap;
    v2f a, b;
    a.x = av.x; a.y = av.y;
    b.x = Bp[0];
    b.y = Bp[Ndim];
    c = __builtin_amdgcn_wmma_f32_16x16x4_f32(false, a, false, b, (short)0, c,
                                              false, false);
    Ap += 4;
    Bp += (size_t)4 * Ndim;
  }

  // ---- tail chunk (K % 4 != 0; only K=14 here): clamp + select, no divergence
  if (Kdim & 3) {
    const int k0 = kfull * 4 + half * 2;
    const int i0 = (k0     < Kdim) ? k0     : Kdim - 1;
    const int i1 = (k0 + 1 < Kdim) ? k0 + 1 : Kdim - 1;
    const float ax = A[(size_t)arow * Kdim + i0];
    const float ay = A[(size_t)arow * Kdim + i1];
    const float bx = B[(size_t)i0 * Ndim + tn*16 + l];
    const float by = B[(size_t)i1 * Ndim + tn*16 + l];
    v2f a, b;
    a.x = (k0     < Kdim) ? ax : 0.f;
    a.y = (k0 + 1 < Kdim) ? ay : 0.f;
    b.x = (k0     < Kdim) ? bx : 0.f;
    b.y = (k0 + 1 < Kdim) ? by : 0.f;
    c = __builtin_amdgcn_wmma_f32_16x16x4_f32(false, a, false, b, (short)0, c,
                                              false, false);
  }

#pragma unroll
  for (int r = 0; r < 8; ++r) Cp[(size_t)r * Ndim] = c[r];
}

// ---------------- misc elementwise ----------------
__global__ void fillk(float* __restrict__ p, float v, int n) {
  int t = blockIdx.x * blockDim.x + threadIdx.x;
  if (t < n) p[t] = v;
}

// x = elu(x + b) over [n, dim] (dim power of two)
__global__ void bias_elu(float* __restrict__ x, const float* __restrict__ b,
                         int n, int dim) {
  int t = blockIdx.x * blockDim.x + threadIdx.x;
  if (t >= n * dim) return;
  float v = x[t] + b[t & (dim - 1)];
  x[t] = v > 0.f ? v : (expf(v) - 1.f);
}

// x = relu(x + b)
__global__ void bias_relu(float* __restrict__ x, const float* __restrict__ b,
                          int n, int dim) {
  int t = blockIdx.x * blockDim.x + threadIdx.x;
  if (t >= n * dim) return;
  float v = x[t] + b[t & (dim - 1)];
  x[t] = v > 0.f ? v : 0.f;
}

// ---------------- GAT attention ----------------
// es[i,h] = sum_c h[i,h,c]*att_s[h,c] ; ed likewise
__global__ void gat_logits(const float* __restrict__ h, const float* __restrict__ as_,
                           const float* __restrict__ ad_, float* __restrict__ es,
                           float* __restrict__ ed, int n, int heads) {
  int t = blockIdx.x * blockDim.x + threadIdx.x;
  if (t >= n * heads) return;
  int i = t / heads, hh = t - i * heads;
  const float* hp = h + (size_t)i * heads * 64 + hh * 64;
  const float* sa = as_ + hh * 64;
  const float* da = ad_ + hh * 64;
  float ss = 0.f, sd = 0.f;
  for (int c = 0; c < 64; ++c) { float v = hp[c]; ss += v * sa[c]; sd += v * da[c]; }
  es[t] = ss; ed[t] = sd;
}

__global__ void edge_max(const int* __restrict__ ei, const float* __restrict__ es,
                         const float* __restrict__ ed, float* __restrict__ m, int heads) {
  int t = blockIdx.x * blockDim.x + threadIdx.x;
  if (t >= ET * heads) return;
  int e = t / heads, hh = t - e * heads;
  int s, d; edge_sd(ei, e, s, d);
  float a = es[s * heads + hh] + ed[d * heads + hh];
  a = a > 0.f ? a : 0.2f * a;                 // leaky_relu(0.2)
  atomicMaxF(&m[d * heads + hh], a);
}

__global__ void edge_expsum(const int* __restrict__ ei, const float* __restrict__ es,
                            const float* __restrict__ ed, const float* __restrict__ m,
                            float* __restrict__ ssum, int heads) {
  int t = blockIdx.x * blockDim.x + threadIdx.x;
  if (t >= ET * heads) return;
  int e = t / heads, hh = t - e * heads;
  int s, d; edge_sd(ei, e, s, d);
  float a = es[s * heads + hh] + ed[d * heads + hh];
  a = a > 0.f ? a : 0.2f * a;
  atomicAdd(&ssum[d * heads + hh], expf(a - m[d * heads + hh]));
}

// GAT1 aggregation: wave per edge, 4 heads x 64 ch; lane -> 8 channels of head lane/8
__global__ void gat_agg4(const int* __restrict__ ei, const float* __restrict__ es,
                         const float* __restrict__ ed, const float* __restrict__ m,
                         const float* __restrict__ ssum, const float* __restrict__ h,
                         float* __restrict__ out) {
  int wid  = (blockIdx.x * blockDim.x + threadIdx.x) >> 5;
  int lane = threadIdx.x & 31;
  if (wid >= ET) return;
  int s, d; edge_sd(ei, wid, s, d);
  int hh = lane >> 3;
  float a = es[s * 4 + hh] + ed[d * 4 + hh];
  a = a > 0.f ? a : 0.2f * a;
  float alpha = expf(a - m[d * 4 + hh]) / (ssum[d * 4 + hh] + 1e-16f);
  int ch = lane * 8;
  const float4* hp = (const float4*)(h + (size_t)s * 256 + ch);
  float4 v0 = hp[0], v1 = hp[1];
  float* op = out + (size_t)d * 256 + ch;
  atomicAdd(op + 0, v0.x * alpha); atomicAdd(op + 1, v0.y * alpha);
  atomicAdd(op + 2, v0.z * alpha); atomicAdd(op + 3, v0.w * alpha);
  atomicAdd(op + 4, v1.x * alpha); atomicAdd(op + 5, v1.y * alpha);
  atomicAdd(op + 6, v1.z * alpha); atomicAdd(op + 7, v1.w * alpha);
}

// GAT2 aggregation: wave per edge, 1 head x 64 ch; lane -> 2 channels
__global__ void gat_agg1(const int* __restrict__ ei, const float* __restrict__ es,
                         const float* __restrict__ ed, const float* __restrict__ m,
                         const float* __restrict__ ssum, const float* __restrict__ h,
                         float* __restrict__ out) {
  int wid  = (blockIdx.x * blockDim.x + threadIdx.x) >> 5;
  int lane = threadIdx.x & 31;
  if (wid >= ET) return;
  int s, d; edge_sd(ei, wid, s, d);
  float a = es[s] + ed[d];
  a = a > 0.f ? a : 0.2f * a;
  float alpha = expf(a - m[d]) / (ssum[d] + 1e-16f);
  int ch = lane * 2;
  const float2 v = *(const float2*)(h + (size_t)s * 64 + ch);
  float* op = out + (size_t)d * 64 + ch;
  atomicAdd(op + 0, v.x * alpha);
  atomicAdd(op + 1, v.y * alpha);
}

// ---------------- GCN ----------------
__global__ void deg_count(const int* __restrict__ ei, float* __restrict__ deg) {
  int e = blockIdx.x * blockDim.x + threadIdx.x;
  if (e >= ET) return;
  int s, d; edge_sd(ei, e, s, d);
  atomicAdd(&deg[d], 1.f);
}

__global__ void deg_inv(const float* __restrict__ deg, float* __restrict__ dinv, int n) {
  int i = blockIdx.x * blockDim.x + threadIdx.x;
  if (i >= n) return;
  float dg = deg[i];
  dinv[i] = dg > 0.f ? 1.f / sqrtf(dg) : 0.f;
}

__global__ void gcn_agg(const int* __restrict__ ei, const float* __restrict__ dinv,
                        const float* __restrict__ xw, float* __restrict__ out) {
  int wid  = (blockIdx.x * blockDim.x + threadIdx.x) >> 5;
  int lane = threadIdx.x & 31;
  if (wid >= ET) return;
  int s, d; edge_sd(ei, wid, s, d);
  float nrm = dinv[s] * dinv[d];
  int ch = lane * 2;
  const float2 v = *(const float2*)(xw + (size_t)s * 64 + ch);
  float* op = out + (size_t)d * 64 + ch;
  atomicAdd(op + 0, nrm * v.x);
  atomicAdd(op + 1, nrm * v.y);
}

// ---------------- pooling + heads ----------------
__global__ void pool_cnt(const int* __restrict__ batch, float* __restrict__ cnt) {
  int i = blockIdx.x * blockDim.x + threadIdx.x;
  if (i >= NN) return;
  atomicAdd(&cnt[batch[i]], 1.f);
}

__global__ void pool_sum(const int* __restrict__ batch, const float* __restrict__ fused,
                         float* __restrict__ pooled) {
  int t = blockIdx.x * blockDim.x + threadIdx.x;
  if (t >= NN * 64) return;
  int i = t >> 6, ch = t & 63;
  atomicAdd(&pooled[batch[i] * 64 + ch], fused[t]);
}

// block per graph, 64 threads: mean pool + 2 MLP heads -> out[0:640] situ, [640:1280] path
__global__ void mlp_heads(const float* __restrict__ pooled, const float* __restrict__ cnt,
                          const float* __restrict__ Wc1, const float* __restrict__ bc1,
                          const float* __restrict__ Wc2, const float* __restrict__ bc2,
                          const float* __restrict__ Wp1, const float* __restrict__ bp1,
                          const float* __restrict__ Wp2, const float* __restrict__ bp2,
                          float* __restrict__ out) {
  __shared__ float prow[64];
  __shared__ float hid[64];        // [0:32) situ hidden, [32:64) path hidden
  int g = blockIdx.x, t = threadIdx.x;
  float cv = cnt[g]; cv = cv > 1.f ? cv : 1.f;
  prow[t] = pooled[g * 64 + t] / cv;
  __syncthreads();
  {
    const float* W = (t < 32) ? Wc1 : Wp1;
    const float* b = (t < 32) ? bc1 : bp1;
    int j = t & 31;
    float acc = b[j];
    for (int c = 0; c < 64; ++c) acc += prow[c] * W[c * 32 + j];
    hid[t] = acc > 0.f ? acc : 0.f;
  }
  __syncthreads();
  if (t < 10) {
    int k = t % 5; bool situ = (t < 5);
    const float* W  = situ ? Wc2 : Wp2;
    const float* b2 = situ ? bc2 : bp2;
    const float* hp = situ ? hid : hid + 32;
    float acc = b2[k];
    for (int j = 0; j < 32; ++j) acc += hp[j] * W[j * 5 + k];
    out[(situ ? 0 : NG * 5) + g * 5 + k] = acc;
  }
}

// ---------------- host launch ----------------
static inline unsigned nb(long long threads) { return (unsigned)((threads + 255) / 256); }

extern "C" void kernel_launch(void* const* d_in, const int* in_sizes, int n_in,
                              void* d_out, int out_size, void* d_ws, size_t ws_size,
                              hipStream_t stream) {
  const float* x   = (const float*)d_in[0];
  const int*   ei  = (const int*)  d_in[1];
  // d_in[2] edge_attr unused by reference
  const int*   bat = (const int*)  d_in[3];
  const float* W1  = (const float*)d_in[4];
  const float* as1 = (const float*)d_in[5];
  const float* ad1 = (const float*)d_in[6];
  const float* b1  = (const float*)d_in[7];
  const float* W2  = (const float*)d_in[8];
  const float* as2 = (const float*)d_in[9];
  const float* ad2 = (const float*)d_in[10];
  const float* b2  = (const float*)d_in[11];
  const float* Wg  = (const float*)d_in[12];
  const float* bg  = (const float*)d_in[13];
  const float* Wf  = (const float*)d_in[14];
  const float* bf  = (const float*)d_in[15];
  const float* Wc1 = (const float*)d_in[16];
  const float* bc1 = (const float*)d_in[17];
  const float* Wc2 = (const float*)d_in[18];
  const float* bc2 = (const float*)d_in[19];
  const float* Wp1 = (const float*)d_in[20];
  const float* bp1 = (const float*)d_in[21];
  const float* Wp2 = (const float*)d_in[22];
  const float* bp2 = (const float*)d_in[23];
  float* out = (float*)d_out;

  // workspace layout (floats)
  float* ws = (float*)d_ws;
  size_t off = 0;
  float* regA  = ws + off; off += (size_t)NN * 256;  // h1; later h2|xw|gcagg|fused (4 x N*64)
  float* regB  = ws + off; off += (size_t)NN * 256;  // agg1/g1; later agg2/g2
  float* es1   = ws + off; off += (size_t)NN * 4;
  float* ed1   = ws + off; off += (size_t)NN * 4;
  float* m1    = ws + off; off += (size_t)NN * 4;
  float* s1    = ws + off; off += (size_t)NN * 4;
  float* es2   = ws + off; off += NN;
  float* ed2   = ws + off; off += NN;
  float* m2    = ws + off; off += NN;
  float* s2    = ws + off; off += NN;
  float* deg   = ws + off; off += NN;
  float* dinv  = ws + off; off += NN;
  float* pool  = ws + off; off += (size_t)NG * 64;
  float* cnt   = ws + off; off += NG;

  float* h1    = regA;
  float* h2    = regA;                 // after h1 is dead
  float* xw    = regA + (size_t)NN * 64;
  float* gcagg = regA + (size_t)NN * 128;
  float* fused = regA + (size_t)NN * 192;

  const dim3 TB(256);

  // --- GAT layer 1 ---
  fillk<<<nb((long long)NN*4), TB, 0, stream>>>(m1, -INFINITY, NN*4);
  fillk<<<nb((long long)NN*4), TB, 0, stream>>>(s1, 0.f, NN*4);
  fillk<<<nb((long long)NN*256), TB, 0, stream>>>(regB, 0.f, NN*256);
  fillk<<<nb(NN), TB, 0, stream>>>(deg, 0.f, NN);

  // h1 = x @ W1   [50000,14]x[14,256]
  gemm_wmma_f32<<<nb((long long)(NN/16)*(256/16)*32), TB, 0, stream>>>(x, W1, h1, NN, 256, 14, 0);
  gat_logits<<<nb((long long)NN*4), TB, 0, stream>>>(h1, as1, ad1, es1, ed1, NN, 4);
  edge_max   <<<nb((long long)ET*4), TB, 0, stream>>>(ei, es1, ed1, m1, 4);
  edge_expsum<<<nb((long long)ET*4), TB, 0, stream>>>(ei, es1, ed1, m1, s1, 4);
  gat_agg4   <<<nb((long long)ET*32), TB, 0, stream>>>(ei, es1, ed1, m1, s1, h1, regB);
  bias_elu   <<<nb((long long)NN*256), TB, 0, stream>>>(regB, b1, NN, 256);   // g1

  // --- GAT layer 2 ---
  // h2 = g1 @ W2  [50000,256]x[256,64]  (h1 dead -> regA reused)
  gemm_wmma_f32<<<nb((long long)(NN/16)*(64/16)*32), TB, 0, stream>>>(regB, W2, h2, NN, 64, 256, 0);
  fillk<<<nb(NN), TB, 0, stream>>>(m2, -INFINITY, NN);
  fillk<<<nb(NN), TB, 0, stream>>>(s2, 0.f, NN);
  gat_logits<<<nb(NN), TB, 0, stream>>>(h2, as2, ad2, es2, ed2, NN, 1);
  edge_max   <<<nb(ET), TB, 0, stream>>>(ei, es2, ed2, m2, 1);
  edge_expsum<<<nb(ET), TB, 0, stream>>>(ei, es2, ed2, m2, s2, 1);
  fillk<<<nb((long long)NN*64), TB, 0, stream>>>(regB, 0.f, NN*64);           // agg2 (g1 dead)
  gat_agg1   <<<nb((long long)ET*32), TB, 0, stream>>>(ei, es2, ed2, m2, s2, h2, regB);
  bias_elu   <<<nb((long long)NN*64), TB, 0, stream>>>(regB, b2, NN, 64);     // g2

  // --- GCN branch ---
  gemm_wmma_f32<<<nb((long long)(NN/16)*(64/16)*32), TB, 0, stream>>>(x, Wg, xw, NN, 64, 14, 0);
  deg_count<<<nb(ET), TB, 0, stream>>>(ei, deg);
  deg_inv  <<<nb(NN), TB, 0, stream>>>(deg, dinv, NN);
  fillk    <<<nb((long long)NN*64), TB, 0, stream>>>(gcagg, 0.f, NN*64);
  gcn_agg  <<<nb((long long)ET*32), TB, 0, stream>>>(ei, dinv, xw, gcagg);
  bias_relu<<<nb((long long)NN*64), TB, 0, stream>>>(gcagg, bg, NN, 64);      // gc

  // --- fuse: fused = relu([g2|gc] @ Wf + bf) = relu(g2@Wf[0:64] + gc@Wf[64:128] + bf)
  gemm_wmma_f32<<<nb((long long)(NN/16)*(64/16)*32), TB, 0, stream>>>(regB,  Wf,        fused, NN, 64, 64, 0);
  gemm_wmma_f32<<<nb((long long)(NN/16)*(64/16)*32), TB, 0, stream>>>(gcagg, Wf + 64*64, fused, NN, 64, 64, 1);
  bias_relu<<<nb((long long)NN*64), TB, 0, stream>>>(fused, bf, NN, 64);

  // --- pool + heads ---
  fillk<<<nb(NG*64), TB, 0, stream>>>(pool, 0.f, NG*64);
  fillk<<<nb(NG),    TB, 0, stream>>>(cnt, 0.f, NG);
  pool_cnt<<<nb(NN), TB, 0, stream>>>(bat, cnt);
  pool_sum<<<nb((long long)NN*64), TB, 0, stream>>>(bat, fused, pool);
  mlp_heads<<<NG, 64, 0, stream>>>(pool, cnt, Wc1, bc1, Wc2, bc2, Wp1, bp1, Wp2, bp2, out);

  (void)in_sizes; (void)n_in; (void)out_size; (void)ws_size;
}